// GaussianPerslayPhi_1614907703769
// MI455X (gfx1250) — compile-verified
//
#include <hip/hip_runtime.h>
#include <math.h>

typedef __attribute__((ext_vector_type(2))) float v2f;
typedef __attribute__((ext_vector_type(8))) float v8f;

#define IMG    64
#define NPTS   128
#define NDIAG  64

// One wave (32 lanes) produces one (diagram, point) 64x64 image as 16
// 16x16 tiles, each computed by a single V_WMMA_F32_16X16X4_F32 rank-1
// outer product (K=0 column of A = fy tile, K=0 row of B = fx tile).
__global__ __launch_bounds__(256) void perslay_gauss_wmma(
    const float* __restrict__ diagrams,   // (64,128,2) f32
    const float* __restrict__ variance,   // scalar f32
    float* __restrict__ out)              // (64,128,64,64,1) f32
{
    const int lane = threadIdx.x & 31;
    const int wid  = (int)((blockIdx.x * blockDim.x + threadIdx.x) >> 5); // 0..8191

    // Per-wave diagram point (uniform across the wave).
    const float p0 = diagrams[wid * 2 + 0];
    const float p1 = diagrams[wid * 2 + 1];
    const float d0 = p0;        // birth
    const float d1 = p1 - p0;   // persistence

    const float var   = variance[0];
    const float k     = 1.0f / (2.0f * var * var);              // 1/(2 sigma^2)
    const float norm  = 1.0f / (2.0f * (float)M_PI * var * var); // 1/(2 pi sigma^2)

    const float lo   = -3.0f;
    const float step = 6.0f / 64.0f;   // (hi-lo)/IMAGE_SIZE

    const int  sub    = lane & 15;
    const bool lowHalf = lane < 16;

    // Separable Gaussian factors, 4 tiles of 16 along each axis.
    // Lanes >=16 carry K=2/3 slots of the WMMA A/B operands and must be 0.
    float fx[4]; // includes normalization factor
    float fy[4];
#pragma unroll
    for (int t = 0; t < 4; ++t) {
        const float c  = lo + step * (float)(t * 16 + sub); // shared x/y coords
        const float dx = d0 - c;
        const float dy = d1 - c;
        const float ex = __expf(-(dx * dx) * k) * norm;
        const float ey = __expf(-(dy * dy) * k);
        fx[t] = lowHalf ? ex : 0.0f;
        fy[t] = lowHalf ? ey : 0.0f;
    }

    const int rbase = (lane >> 4) * 8; // D rows: lanes 0-15 -> M=v, 16-31 -> M=v+8
    float* const img = out + (size_t)wid * (IMG * IMG);

#pragma unroll
    for (int ti = 0; ti < 4; ++ti) {
        v2f a;
        a.x = fy[ti];   // K=0 column of A (lanes 0-15, M = sub)
        a.y = 0.0f;     // K=1 (and K=3 in high lanes) zero
#pragma unroll
        for (int tj = 0; tj < 4; ++tj) {
            v2f b;
            b.x = fx[tj];   // K=0 row of B (lanes 0-15, N = sub)
            b.y = 0.0f;
            v8f c = {};
            // D[M][N] = fy[ti*16+M] * fx[tj*16+N]
            v8f d = __builtin_amdgcn_wmma_f32_16x16x4_f32(
                false, a, false, b, (short)0, c, false, false);

            float* o = img + (size_t)(ti * 16) * IMG + (tj * 16);
#pragma unroll
            for (int v = 0; v < 8; ++v) {
                // lanes 0-15 store row M=v cols 0-15 (64B contiguous),
                // lanes 16-31 store row M=v+8 cols 0-15 (64B contiguous)
                o[(size_t)(rbase + v) * IMG + sub] = d[v];
            }
        }
    }
}

extern "C" void kernel_launch(void* const* d_in, const int* in_sizes, int n_in,
                              void* d_out, int out_size, void* d_ws, size_t ws_size,
                              hipStream_t stream) {
    (void)in_sizes; (void)n_in; (void)out_size; (void)d_ws; (void)ws_size;
    const float* diagrams = (const float*)d_in[0];
    const float* variance = (const float*)d_in[1];
    float* out = (float*)d_out;

    // 8192 waves (one per (diagram, point)), 8 waves per 256-thread block.
    const int totalThreads = NDIAG * NPTS * 32;  // 262144
    const int block = 256;
    const int grid = totalThreads / block;       // 1024
    perslay_gauss_wmma<<<grid, block, 0, stream>>>(diagrams, variance, out);
}